// DualGrainSeparatePermuter_65901978190084
// MI455X (gfx1250) — compile-verified
//
#include <hip/hip_runtime.h>
#include <hip/hip_bf16.h>

// DualGrainSeparatePermuter for MI455X (gfx1250).
// Pure data-movement kernel: per-row stream compaction + EOS/PAD fill.
// Strategy: 1 workgroup (256 thr = 8 wave32) per batch row.
//   - async global->LDS staging of the 4KB indices row (CDNA5 ASYNCcnt path)
//   - wave32 ballot + popc exclusive prefix sums for compaction slots
//   - direct coalesced global stores for scatter / EOS / PAD / segments

#define NREG   256
#define CLEN   257     // coarse stream length (256 + 1)
#define FLEN   1025    // fine stream length (1024 + 1)
#define CONTENT_PAD 1024.0f
#define CONTENT_EOS 1025.0f
#define CPOS_PAD    256.0f
#define CPOS_EOS    257.0f
#define FPOS_PAD    1024.0f
#define FPOS_EOS    1025.0f

#if defined(__HIP_DEVICE_COMPILE__)
#if __has_builtin(__builtin_amdgcn_global_load_async_to_lds_b128)
#define HAVE_ASYNC_LDS 1
#endif
#if __has_builtin(__builtin_amdgcn_ballot_w32)
#define HAVE_BALLOT32 1
#endif
#endif

// 16-byte int vector + address-space-qualified pointee types for the async builtin
typedef int v4i __attribute__((vector_size(16)));
typedef __attribute__((address_space(1))) v4i g_v4i;   // global
typedef __attribute__((address_space(3))) v4i l_v4i;   // LDS

static __device__ __forceinline__ unsigned ballot32(bool p) {
#ifdef HAVE_BALLOT32
    return __builtin_amdgcn_ballot_w32(p);
#else
    return (unsigned)__ballot(p);   // wave32: low 32 bits valid
#endif
}

__global__ __launch_bounds__(256) void dgsp_kernel(
    const int* __restrict__ indices,      // [B, 32, 32]
    const int* __restrict__ grain,        // [B, 16, 16] (values 0/1)
    float* __restrict__ out,              // concat of 6 outputs, flat
    int B)
{
    const int b = blockIdx.x;
    const int t = threadIdx.x;            // region id 0..255

    __shared__ int lds_idx[1024];         // staged indices row (4KB)
    __shared__ int cwt[8];                // per-wave coarse totals
    __shared__ int fwt[8];                // per-wave fine totals

    const int* row = indices + (size_t)b * 1024;

    // ---- stage the 4KB indices row into LDS (async DMA on CDNA5) ----
#ifdef HAVE_ASYNC_LDS
    __builtin_amdgcn_global_load_async_to_lds_b128(
        (g_v4i*)(row + t * 4), (l_v4i*)&lds_idx[t * 4], 0, 0);
#else
    reinterpret_cast<int4*>(lds_idx)[t] =
        reinterpret_cast<const int4*>(row)[t];
#endif

    // ---- per-region select + wave32 ballot scan (overlaps async DMA) ----
    const int g = grain[(size_t)b * NREG + t];   // 0 => coarse, 1 => fine
    const bool csel = (g == 0);

    const unsigned lane = (unsigned)t & 31u;
    const unsigned wave = (unsigned)t >> 5;

    const unsigned cm = ballot32(csel);
    const unsigned fm = ballot32(!csel);
    const unsigned lmask = (lane == 0) ? 0u : (0xFFFFFFFFu >> (32u - lane));
    const int cpre_w = __popc(cm & lmask);
    const int fpre_w = __popc(fm & lmask);
    if (lane == 31) { cwt[wave] = __popc(cm); fwt[wave] = __popc(fm); }

    // wait for this wave's async transfer, then cross-wave barrier
#ifdef HAVE_ASYNC_LDS
#if __has_builtin(__builtin_amdgcn_s_wait_asynccnt)
    __builtin_amdgcn_s_wait_asynccnt(0);
#else
    asm volatile("s_wait_asynccnt 0" ::: "memory");
#endif
#endif
    __syncthreads();

    int cbase = 0, fbase = 0, ctot = 0, ftot = 0;
    #pragma unroll
    for (int w = 0; w < 8; ++w) {
        const int c = cwt[w], f = fwt[w];
        if (w < (int)wave) { cbase += c; fbase += f; }
        ctot += c; ftot += f;
    }
    const int cslot  = cbase + cpre_w;   // exclusive prefix among coarse-selected
    const int fslot  = 4 * (fbase + fpre_w);
    const int ccount = ctot;             // # coarse tokens
    const int fcount = 4 * ftot;         // # fine tokens

    // ---- gather this region's 2x2 sub-tile from LDS ----
    const int h1 = t >> 4, w1 = t & 15;
    const int base = 64 * h1 + 2 * w1;   // == fine position of token 0
    const int c0 = lds_idx[base];
    const int c1 = lds_idx[base + 1];
    const int c2 = lds_idx[base + 32];
    const int c3 = lds_idx[base + 33];

    // ---- output row pointers (6 concatenated arrays) ----
    const size_t Bz = (size_t)B;
    float* cc = out + (size_t)b * CLEN;
    float* fc = out + Bz * CLEN                          + (size_t)b * FLEN;
    float* cp = out + Bz * (CLEN + FLEN)                 + (size_t)b * CLEN;
    float* fp = out + Bz * (2 * CLEN + FLEN)             + (size_t)b * FLEN;
    float* cs = out + Bz * (2 * CLEN + 2 * FLEN)         + (size_t)b * CLEN;
    float* fs = out + Bz * (3 * CLEN + 2 * FLEN)         + (size_t)b * FLEN;

    // ---- scatter compacted tokens (slots are monotonic in t -> coalesced-ish) ----
    if (csel) {
        cc[cslot] = (float)c0;
        cp[cslot] = (float)t;
    } else {
        fc[fslot + 0] = (float)c0;
        fc[fslot + 1] = (float)c1;
        fc[fslot + 2] = (float)c2;
        fc[fslot + 3] = (float)c3;
        fp[fslot + 0] = (float)base;
        fp[fslot + 1] = (float)(base + 1);
        fp[fslot + 2] = (float)(base + 32);
        fp[fslot + 3] = (float)(base + 33);
    }

    // ---- EOS / PAD fill + segment constants (covers every output element) ----
    for (int s = t; s < CLEN; s += 256) {
        if (s > ccount)       { cc[s] = CONTENT_PAD; cp[s] = CPOS_PAD; }
        else if (s == ccount) { cc[s] = CONTENT_EOS; cp[s] = CPOS_EOS; }
        cs[s] = 0.0f;
    }
    for (int s = t; s < FLEN; s += 256) {
        if (s > fcount)       { fc[s] = CONTENT_PAD; fp[s] = FPOS_PAD; }
        else if (s == fcount) { fc[s] = CONTENT_EOS; fp[s] = FPOS_EOS; }
        fs[s] = 1.0f;
    }
}

extern "C" void kernel_launch(void* const* d_in, const int* in_sizes, int n_in,
                              void* d_out, int out_size, void* d_ws, size_t ws_size,
                              hipStream_t stream) {
    const int* indices = (const int*)d_in[0];   // [B,32,32]
    const int* grain   = (const int*)d_in[1];   // [B,16,16]
    float* out = (float*)d_out;
    const int B = in_sizes[0] / 1024;

    dgsp_kernel<<<dim3(B), dim3(256), 0, stream>>>(indices, grain, out, B);
}